// GCN_layer_68693706932433
// MI455X (gfx1250) — compile-verified
//
#include <hip/hip_runtime.h>

#define DF 64          // feature dim (reference: 64)
#define BLK 256        // 8 wave32 per block

// ---------------------------------------------------------------- init
__global__ void gcn_init_kernel(float* __restrict__ dinv, int* __restrict__ counts, int n) {
    int i = blockIdx.x * blockDim.x + threadIdx.x;
    if (i < n) { dinv[i] = 0.0f; counts[i] = 0; }
}

// ------------------------------------------------- weighted degree + edge counts
__global__ void gcn_degree_kernel(const float* __restrict__ vals,
                                  const int* __restrict__ rows,
                                  float* __restrict__ dinv,
                                  int* __restrict__ counts, int e) {
    int i = blockIdx.x * blockDim.x + threadIdx.x;
    if (i < e) {
        int r = rows[i];
        atomicAdd(&dinv[r], vals[i]);   // global_atomic_add_f32 (no-return path)
        atomicAdd(&counts[r], 1);
    }
}

// ---------------------------------------------------------------- d^{-1/2}
__global__ void gcn_rsqrt_kernel(float* __restrict__ dinv, int n) {
    int i = blockIdx.x * blockDim.x + threadIdx.x;
    if (i < n) {
        float s = dinv[i];
        dinv[i] = (s > 0.0f) ? rsqrtf(s) : 0.0f;
    }
}

// ------------------------------------- single-block exclusive scan (n up to ~1e6)
__global__ void gcn_scan_kernel(const int* __restrict__ counts,
                                int* __restrict__ row_start, int n) {
    __shared__ int tmp[1024];
    __shared__ int carry;
    if (threadIdx.x == 0) carry = 0;
    __syncthreads();
    for (int base = 0; base < n; base += 1024) {
        int i = base + (int)threadIdx.x;
        int v = (i < n) ? counts[i] : 0;
        tmp[threadIdx.x] = v;
        __syncthreads();
        // Hillis-Steele inclusive scan over the chunk
        for (int off = 1; off < 1024; off <<= 1) {
            int t = (threadIdx.x >= (unsigned)off) ? tmp[threadIdx.x - off] : 0;
            __syncthreads();
            tmp[threadIdx.x] += t;
            __syncthreads();
        }
        int incl = tmp[threadIdx.x];
        int excl = incl - v;
        if (i < n) row_start[i] = carry + excl;
        __syncthreads();                       // everyone done reading carry
        if (threadIdx.x == 1023) carry += incl;
        __syncthreads();
    }
}

// ------------------------------------------------- CSR fill with normalized weights
// Uses row_start as the fill cursor: after this kernel row_start[r] holds the
// INCLUSIVE prefix, so row r occupies [row_start[r-1], row_start[r]).
__global__ void gcn_fill_kernel(const float* __restrict__ vals,
                                const int* __restrict__ rows,
                                const int* __restrict__ cols,
                                const float* __restrict__ dinv,
                                int* __restrict__ row_start,
                                int* __restrict__ csr_col,
                                float* __restrict__ csr_w, int e) {
    int i = blockIdx.x * blockDim.x + threadIdx.x;
    if (i < e) {
        int r = rows[i], c = cols[i];
        float w = dinv[r] * vals[i] * dinv[c];
        int pos = atomicAdd(&row_start[r], 1);
        csr_col[pos] = c;
        csr_w[pos]   = w;
    }
}

// ------------------------------------------- features -> out via async LDS DMA
// CDNA5 path: global_load_async_to_lds_b128 / global_store_async_from_lds_b128
// (ASYNCcnt-tracked DMA, no VGPR staging). Each lane owns a private 16B LDS slot,
// so a per-wave s_wait_asynccnt fully orders the load before the store issues.
// n16 is an exact multiple of BLK for this problem, so no partial waves.
__global__ void gcn_copy_async_kernel(const float* __restrict__ src,
                                      float* __restrict__ dst, int n16) {
    __shared__ __align__(16) char lds_buf[BLK * 16];
    int i = blockIdx.x * blockDim.x + threadIdx.x;
    if (i >= n16) return;
    unsigned lds = (unsigned)(size_t)&lds_buf[threadIdx.x * 16];
    unsigned long long gs = (unsigned long long)(const char*)src + (unsigned long long)i * 16;
    unsigned long long gd = (unsigned long long)(char*)dst + (unsigned long long)i * 16;
    asm volatile("global_load_async_to_lds_b128 %0, %1, off"
                 :: "v"(lds), "v"(gs) : "memory");
    asm volatile("s_wait_asynccnt 0" ::: "memory");
    asm volatile("global_store_async_from_lds_b128 %0, %1, off"
                 :: "v"(gd), "v"(lds) : "memory");
    // s_endpgm performs an implicit wait-idle, covering the outstanding store
}

// ------------------------------------------------- SpMM: one wave32 per row
// lane l owns dims [2l, 2l+1]  (64 floats = 32 lanes x float2, b64 loads/stores).
// All row-level values are wave-uniform: readfirstlane pins them to SGPRs so the
// gathers become saddr-form global_load_b64 and loop bookkeeping rides the SALU.
__global__ void gcn_spmm_kernel(const float* __restrict__ features,
                                const int* __restrict__ row_start,
                                const int* __restrict__ csr_col,
                                const float* __restrict__ csr_w,
                                const int* __restrict__ index,
                                float* __restrict__ out, int n) {
    int wave = (int)((blockIdx.x * blockDim.x + threadIdx.x) >> 5);
    int lane = (int)(threadIdx.x & 31u);
    if (wave >= n) return;
    int r = wave;
    int begin = (r == 0) ? 0 : row_start[r - 1];
    int end   = row_start[r];
    begin = __builtin_amdgcn_readfirstlane(begin);
    end   = __builtin_amdgcn_readfirstlane(end);

    const float* fb = features + 2 * lane;   // this lane's slice within any row
    float2 acc; acc.x = 0.0f; acc.y = 0.0f;

    for (int e = begin; e < end; ++e) {
        int   c = __builtin_amdgcn_readfirstlane(csr_col[e]);
        float w = __builtin_amdgcn_readfirstlane(csr_w[e]);
        if (e + 1 < end) {
            int cn = __builtin_amdgcn_readfirstlane(csr_col[e + 1]);
            // gfx1250: global_prefetch_b8 — hide latency of the next random gather
            __builtin_prefetch(fb + (size_t)cn * DF, 0, 0);
        }
        float2 f = *(const float2*)(fb + (size_t)c * DF);
        acc.x = fmaf(w, f.x, acc.x);
        acc.y = fmaf(w, f.y, acc.y);
    }

    int orow = __builtin_amdgcn_readfirstlane(index[r]);  // result[index[r]] = row
    *(float2*)(out + (size_t)orow * DF + 2 * lane) = acc;
}

// ================================================================ launch
extern "C" void kernel_launch(void* const* d_in, const int* in_sizes, int n_in,
                              void* d_out, int out_size, void* d_ws, size_t ws_size,
                              hipStream_t stream) {
    (void)n_in; (void)out_size; (void)ws_size;

    const float* features  = (const float*)d_in[0];
    const float* edge_vals = (const float*)d_in[1];
    const int*   edge_rows = (const int*)d_in[2];
    const int*   edge_cols = (const int*)d_in[3];
    const int*   index     = (const int*)d_in[4];
    float*       out       = (float*)d_out;

    const int E = in_sizes[1];   // edges
    const int N = in_sizes[4];   // nodes (index covers all rows)

    // workspace carve-out (256B aligned slices): ~14.1 MB total
    char*  ws  = (char*)d_ws;
    size_t off = 0;
    auto take = [&](size_t bytes) -> char* {
        char* p = ws + off;
        off = (off + bytes + 255) & ~(size_t)255;
        return p;
    };
    float* dinv      = (float*)take((size_t)N * sizeof(float));
    int*   counts    = (int*)  take((size_t)N * sizeof(int));
    int*   row_start = (int*)  take((size_t)N * sizeof(int));
    int*   csr_col   = (int*)  take((size_t)E * sizeof(int));
    float* csr_w     = (float*)take((size_t)E * sizeof(float));

    const int gN = (N + BLK - 1) / BLK;
    const int gE = (E + BLK - 1) / BLK;

    gcn_init_kernel  <<<gN, BLK, 0, stream>>>(dinv, counts, N);
    gcn_degree_kernel<<<gE, BLK, 0, stream>>>(edge_vals, edge_rows, dinv, counts, E);
    gcn_rsqrt_kernel <<<gN, BLK, 0, stream>>>(dinv, N);
    gcn_scan_kernel  <<<1, 1024, 0, stream>>>(counts, row_start, N);
    gcn_fill_kernel  <<<gE, BLK, 0, stream>>>(edge_vals, edge_rows, edge_cols,
                                              dinv, row_start, csr_col, csr_w, E);

    const int n16 = (N * DF) / 4;   // 16-byte packets
    gcn_copy_async_kernel<<<(n16 + BLK - 1) / BLK, BLK, 0, stream>>>(features, out, n16);

    const int rows_per_block = BLK / 32;     // wave32: 8 rows per 256-thread block
    gcn_spmm_kernel  <<<(N + rows_per_block - 1) / rows_per_block, BLK, 0, stream>>>(
        features, row_start, csr_col, csr_w, index, out, N);
}